// BondConstraintLayer_33887291965649
// MI455X (gfx1250) — compile-verified
//
#include <hip/hip_runtime.h>
#include <stdint.h>

// BondConstraintLayer for MI455X (gfx1250).
// Memory-bound streaming kernel: ~154 MB HBM traffic -> ~6.6 us floor @ 23.3 TB/s.
// CDNA5 paths: GLOBAL_LOAD_ASYNC_TO_LDS_B128 staging (ASYNCcnt) + s_wait_asynccnt,
// b128 non-temporal stores, wave32 LDS reduction, deterministic two-pass reduction.
// 32-bit index math (E*5 = 16M < 2^31) and a block-uniform full-tile fast path
// keep the hot loop straight-line with no EXEC churn.

#define BLOCK 256
#define TILE_EDGES 1024
#define TILE_FLOATS (TILE_EDGES * 5)              // 5120 floats = 20 KB LDS tile
#define VEC_PER_THREAD (TILE_FLOATS / 4 / BLOCK)  // 5 b128 transfers per thread

// Native clang vector type: accepted by __builtin_nontemporal_store
// (HIP's float4 is a struct and is rejected).
typedef float f32x4 __attribute__((ext_vector_type(4)));

// Async copy of 16 bytes global -> LDS. VDST holds the LDS byte address
// (flat shared pointer truncated to 32 bits == DS-space offset, ISA 10.2).
__device__ __forceinline__ void async_load_b128(uint32_t lds_byte_addr, const void* gaddr) {
  asm volatile("global_load_async_to_lds_b128 %0, %1, off"
               :: "v"(lds_byte_addr), "v"((uint64_t)(uintptr_t)gaddr)
               : "memory");
}

__device__ __forceinline__ void wait_async0() {
  asm volatile("s_wait_asynccnt 0" ::: "memory");
}

__device__ __forceinline__ float edge_update(
    float* tile, int le, int e, int E,
    const int* __restrict__ eidx, const int* __restrict__ atom_types) {
  const int r  = eidx[e];
  const int c  = eidx[E + e];
  const int t1 = atom_types[r];
  const int t2 = atom_types[c];
  const float m4 = (t1 == 4 || t2 == 4) ? 1.0f : 0.0f;
  const float m5 = (t1 == 5 || t2 == 5) ? 1.0f : 0.0f;
  const float p4 = 100.0f * m4;
  const float p5 = 50.0f * m5;
  const int b = le * 5;
  const float a1 = tile[b + 1] - p4;
  const float a2 = tile[b + 2] - p4;
  const float a3 = tile[b + 3] - p4;
  const float a4 = tile[b + 4] - p4;
  // col 0 untouched; col 1 only -100*m4; cols 2..4 also -50*m5
  tile[b + 1] = a1;
  tile[b + 2] = a2 - p5;
  tile[b + 3] = a3 - p5;
  tile[b + 4] = a4 - p5;
  // violations measured AFTER -100 step, BEFORE -50 step (cols 1..4)
  return m4 * (fmaxf(a1, 0.0f) + fmaxf(a2, 0.0f) +
               fmaxf(a3, 0.0f) + fmaxf(a4, 0.0f));
}

__global__ __launch_bounds__(BLOCK) void bond_constraint_33887_main(
    const float* __restrict__ logits,      // [E,5]
    const int* __restrict__ eidx,          // [2,E] flat
    const int* __restrict__ atom_types,    // [N]
    float* __restrict__ out,               // [E,5]
    float* __restrict__ partial,           // [gridDim.x]
    int E) {
  __shared__ __align__(16) float tile[TILE_FLOATS];
  __shared__ float red[BLOCK];

  const int tid = threadIdx.x;
  const int e0 = (int)blockIdx.x * TILE_EDGES;
  const int f0 = e0 * 5;        // global float index of tile start (< 2^31)
  const int nf = E * 5;

  float v = 0.0f;

  if (e0 + TILE_EDGES <= E) {
    // ================= block-uniform FAST path: full tile =================
    // Stage 20 KB of logits into LDS with 1280 async b128 copies.
    #pragma unroll
    for (int k = 0; k < VEC_PER_THREAD; ++k) {
      const int v4 = tid + k * BLOCK;
      async_load_b128((uint32_t)(uintptr_t)&tile[v4 * 4], logits + f0 + v4 * 4);
    }
    wait_async0();
    __syncthreads();

    // 4 edges per thread, stride BLOCK (bank-conflict-free: gcd(5,64)==1).
    #pragma unroll
    for (int j = 0; j < TILE_EDGES / BLOCK; ++j) {
      const int le = tid + j * BLOCK;
      v += edge_update(tile, le, e0 + le, E, eidx, atom_types);
    }
    __syncthreads();

    // Stream tile out with aligned non-temporal b128 stores.
    #pragma unroll
    for (int k = 0; k < VEC_PER_THREAD; ++k) {
      const int v4 = tid + k * BLOCK;
      const f32x4 val = ((const f32x4*)tile)[v4];
      __builtin_nontemporal_store(val, (f32x4*)(out + f0 + v4 * 4));
    }
  } else {
    // ================= block-uniform SLOW path: partial tile ==============
    #pragma unroll
    for (int k = 0; k < VEC_PER_THREAD; ++k) {
      const int v4 = tid + k * BLOCK;
      const int g = f0 + v4 * 4;
      #pragma unroll
      for (int j = 0; j < 4; ++j)
        tile[v4 * 4 + j] = (g + j < nf) ? logits[g + j] : 0.0f;
    }
    __syncthreads();

    #pragma unroll
    for (int j = 0; j < TILE_EDGES / BLOCK; ++j) {
      const int le = tid + j * BLOCK;
      const int e = e0 + le;
      if (e < E) v += edge_update(tile, le, e, E, eidx, atom_types);
    }
    __syncthreads();

    #pragma unroll
    for (int k = 0; k < VEC_PER_THREAD; ++k) {
      const int v4 = tid + k * BLOCK;
      const int g = f0 + v4 * 4;
      for (int j = 0; j < 4; ++j)
        if (g + j < nf) out[g + j] = tile[v4 * 4 + j];
    }
  }

  // ---- Deterministic block reduction of violation partials.
  red[tid] = v;
  __syncthreads();
  #pragma unroll
  for (int o = BLOCK / 2; o > 0; o >>= 1) {
    if (tid < o) red[tid] += red[tid + o];
    __syncthreads();
  }
  if (tid == 0) partial[blockIdx.x] = red[0];
}

__global__ __launch_bounds__(BLOCK) void bond_constraint_33887_reduce(
    const float* __restrict__ partial, int n, float* __restrict__ dst, float scale) {
  __shared__ float red[BLOCK];
  float s = 0.0f;
  for (int i = threadIdx.x; i < n; i += BLOCK) s += partial[i];
  red[threadIdx.x] = s;
  __syncthreads();
  #pragma unroll
  for (int o = BLOCK / 2; o > 0; o >>= 1) {
    if (threadIdx.x < o) red[threadIdx.x] += red[threadIdx.x + o];
    __syncthreads();
  }
  if (threadIdx.x == 0) dst[0] = red[0] * scale;
}

extern "C" void kernel_launch(void* const* d_in, const int* in_sizes, int n_in,
                              void* d_out, int out_size, void* d_ws, size_t ws_size,
                              hipStream_t stream) {
  const float* logits     = (const float*)d_in[0];  // [E,5] f32
  const int*   eidx       = (const int*)d_in[1];    // [2,E] i32
  const int*   atom_types = (const int*)d_in[2];    // [N]   i32
  float*       out        = (float*)d_out;          // [E*5] + 1 scalar

  const int E = in_sizes[0] / 5;
  const int nBlocks = (E + TILE_EDGES - 1) / TILE_EDGES;
  float* partial = (float*)d_ws;  // nBlocks floats of scratch

  bond_constraint_33887_main<<<nBlocks, BLOCK, 0, stream>>>(
      logits, eidx, atom_types, out, partial, E);
  bond_constraint_33887_reduce<<<1, BLOCK, 0, stream>>>(
      partial, nBlocks, out + (long)E * 5, 1.0f / (float)E);
}